// GIN_23536420782703
// MI455X (gfx1250) — compile-verified
//
#include <hip/hip_runtime.h>
#include <hip/hip_bf16.h>

#define H 128
#define NEG_SLOPE 0.01f
#define BN_EPS 1e-5f

typedef float v2f __attribute__((ext_vector_type(2)));
typedef float v8f __attribute__((ext_vector_type(8)));

// ---------------------------------------------------------------------------
// WMMA fp32 GEMM: out[Nrows,128] = act(A @ W + bias)   (Nrows % 16 == 0)
//   A_T == false : A row-major [Nrows, K]
//   A_T == true  : A stored [K, Nrows] (the input x), lane-coalesced loads
// One 256-thread block (8 waves) computes a 16-row stripe; wave w owns
// columns [16w, 16w+16). K-loop issues V_WMMA_F32_16X16X4_F32 per 4 k's.
// Optional out2: duplicate the result into a second buffer (agg init).
// ---------------------------------------------------------------------------
template <int K, bool LEAKY, bool A_T>
__global__ __launch_bounds__(256) void gemm_wmma_kernel(
    const float* __restrict__ A, const float* __restrict__ W,
    const float* __restrict__ bias, float* __restrict__ out,
    float* __restrict__ out2, int Nrows) {
  const int lane = threadIdx.x & 31;
  const int wave = threadIdx.x >> 5;
  const int hi   = lane >> 4;    // lane>=16 half holds K+2 / K+3 of fragments
  const int l16  = lane & 15;
  const int mtile = blockIdx.x * 16;
  const int jbase = wave * 16;   // 8 waves cover all 128 output columns
  const int m = mtile + l16;     // A-fragment row for this lane

  v8f acc = {};
#pragma unroll 4
  for (int k = 0; k < K; k += 4) {
    const int ka = k + 2 * hi;
    v2f a, b;
    if (A_T) {  // A[k][n] layout (x is [F_IN, N])
      a.x = A[(size_t)ka * Nrows + m];
      a.y = A[(size_t)(ka + 1) * Nrows + m];
    } else {    // A[n][k] layout, 8B vector load
      a = *(const v2f*)(A + (size_t)m * K + ka);
    }
    // B fragment: VGPR0 = row (k+2*hi), VGPR1 = row (k+2*hi+1), col = l16
    b.x = W[ka * H + jbase + l16];
    b.y = W[(ka + 1) * H + jbase + l16];
    acc = __builtin_amdgcn_wmma_f32_16x16x4_f32(
        /*neg_a=*/false, a, /*neg_b=*/false, b,
        /*c_mod=*/(short)0, acc, /*reuse_a=*/false, /*reuse_b=*/false);
  }

  const int col = jbase + l16;
  const float bb = bias[col];
  float res[8];
#pragma unroll
  for (int r = 0; r < 8; ++r) {  // D: VGPR r -> row r (lanes<16) / r+8 (>=16)
    float v = acc[r] + bb;
    if (LEAKY) v = v > 0.f ? v : NEG_SLOPE * v;
    res[r] = v;
  }
#pragma unroll
  for (int r = 0; r < 8; ++r)
    out[(size_t)(mtile + r + 8 * hi) * H + col] = res[r];
  if (out2) {
#pragma unroll
    for (int r = 0; r < 8; ++r)
      out2[(size_t)(mtile + r + 8 * hi) * H + col] = res[r];
  }
}

// ---------------------------------------------------------------------------
// agg[dst[e], :] += h[src[e], :]   — one wave32 per edge, float4 per lane,
// hardware global_atomic_add_f32 (node table is L2-resident; 192MB L2).
// ---------------------------------------------------------------------------
__global__ __launch_bounds__(256) void edge_scatter_kernel(
    const float* __restrict__ h, const long long* __restrict__ src,
    const long long* __restrict__ dst, float* __restrict__ agg, int E) {
  const int lane = threadIdx.x & 31;
  int e = (int)(((size_t)blockIdx.x * blockDim.x + threadIdx.x) >> 5);
  const int nwaves = (int)(((size_t)gridDim.x * blockDim.x) >> 5);
  for (; e < E; e += nwaves) {
    const long long s = src[e];
    const long long d = dst[e];
    const float4 v = *(const float4*)(h + (size_t)s * H + lane * 4);
    float* p = agg + (size_t)d * H + lane * 4;
    unsafeAtomicAdd(p + 0, v.x);
    unsafeAtomicAdd(p + 1, v.y);
    unsafeAtomicAdd(p + 2, v.z);
    unsafeAtomicAdd(p + 3, v.w);
  }
}

// ---------------------------------------------------------------------------
// BatchNorm (training mode, biased variance) over node axis
// ---------------------------------------------------------------------------
__global__ void zero_kernel(float* p, int n) {
  int i = blockIdx.x * blockDim.x + threadIdx.x;
  if (i < n) p[i] = 0.f;
}

__global__ __launch_bounds__(128) void bn_reduce_kernel(
    const float* __restrict__ h, float* __restrict__ stats, int Nrows) {
  const int c = threadIdx.x;  // channel 0..127, coalesced across lanes
  float s = 0.f, ss = 0.f;
  for (int r = blockIdx.x; r < Nrows; r += gridDim.x) {
    const float v = h[(size_t)r * H + c];
    s += v;
    ss += v * v;
  }
  unsafeAtomicAdd(&stats[c], s);
  unsafeAtomicAdd(&stats[H + c], ss);
}

// Normalizes h in place AND duplicates the result into h2 (next layer's agg
// init) — removes the separate 51MB copy pass per layer.
__global__ __launch_bounds__(256) void bn_apply_kernel(
    float* __restrict__ h, float* __restrict__ h2,
    const float* __restrict__ stats, const float* __restrict__ gamma,
    const float* __restrict__ beta, int Nrows) {
  size_t i = (size_t)blockIdx.x * blockDim.x + threadIdx.x;
  const size_t total = (size_t)Nrows * H;
  const size_t stride = (size_t)gridDim.x * blockDim.x;
  const float invN = 1.0f / (float)Nrows;
  for (; i < total; i += stride) {
    const int c = (int)(i & (H - 1));
    const float mean = stats[c] * invN;
    const float var = stats[H + c] * invN - mean * mean;
    const float scale = rsqrtf(var + BN_EPS) * gamma[c];
    const float v = (h[i] - mean) * scale + beta[c];
    h[i] = v;
    h2[i] = v;
  }
}

// ---------------------------------------------------------------------------
// out[N,8] = t[N,128] @ W2[128,8] + b2   (tiny tail GEMM, VALU is fine)
// ---------------------------------------------------------------------------
__global__ __launch_bounds__(256) void post2_kernel(
    const float* __restrict__ t, const float* __restrict__ W2,
    const float* __restrict__ b2, float* __restrict__ out, int Nrows) {
  const int idx = blockIdx.x * blockDim.x + threadIdx.x;
  if (idx >= Nrows * 8) return;
  const int n = idx >> 3, o = idx & 7;
  float acc = b2[o];
#pragma unroll 8
  for (int k = 0; k < H; ++k) acc += t[(size_t)n * H + k] * W2[k * 8 + o];
  out[idx] = acc;
}

// ---------------------------------------------------------------------------
extern "C" void kernel_launch(void* const* d_in, const int* in_sizes, int n_in,
                              void* d_out, int out_size, void* d_ws,
                              size_t ws_size, hipStream_t stream) {
  const float* x        = (const float*)d_in[0];   // [64, N]
  const long long* ei   = (const long long*)d_in[1];
  const float* pre_W    = (const float*)d_in[2];   // [64, 128]
  const float* pre_b    = (const float*)d_in[3];
  const float* conv_W1  = (const float*)d_in[4];   // [3,128,128]
  const float* conv_b1  = (const float*)d_in[5];
  const float* conv_W2  = (const float*)d_in[6];
  const float* conv_b2  = (const float*)d_in[7];
  const float* bn_g     = (const float*)d_in[8];   // [2,128]
  const float* bn_b     = (const float*)d_in[9];
  const float* post_W1  = (const float*)d_in[10];
  const float* post_b1  = (const float*)d_in[11];
  const float* post_W2  = (const float*)d_in[12];  // [128, 8]
  const float* post_b2  = (const float*)d_in[13];
  float* out = (float*)d_out;

  const int N = in_sizes[0] / 64;  // F_IN = 64; N = 50000 (multiple of 16)
  const int E = in_sizes[1] / 2;
  const long long* src = ei;
  const long long* dst = ei + E;

  float* bufA  = (float*)d_ws;                 // h       [N,H]
  float* bufB  = bufA + (size_t)N * H;         // agg     [N,H]
  float* bufC  = bufB + (size_t)N * H;         // mlp tmp [N,H]
  float* stats = bufC + (size_t)N * H;         // 256 floats (sum | sumsq)

  const int mtiles = (N + 15) / 16;
  const int edgeBlocks = (E + 7) / 8;          // 8 waves/block, wave per edge
  const int applyBlocks = (int)(((size_t)N * H + 255) / 256);

  // pre_mp: h = x^T @ pre_W + pre_b ; dual-write into agg for layer 0
  gemm_wmma_kernel<64, false, true>
      <<<mtiles, 256, 0, stream>>>(x, pre_W, pre_b, bufA, bufB, N);

  for (int l = 0; l < 3; ++l) {
    // GIN aggregation: agg (already = h) += scatter_sum(h[src] -> dst)
    edge_scatter_kernel<<<edgeBlocks, 256, 0, stream>>>(bufA, src, dst, bufB, E);
    // MLP: leaky(agg@W1+b1) @ W2 + b2
    gemm_wmma_kernel<128, true, false><<<mtiles, 256, 0, stream>>>(
        bufB, conv_W1 + (size_t)l * H * H, conv_b1 + l * H, bufC, nullptr, N);
    gemm_wmma_kernel<128, false, false><<<mtiles, 256, 0, stream>>>(
        bufC, conv_W2 + (size_t)l * H * H, conv_b2 + l * H, bufA, nullptr, N);
    if (l < 2) {
      zero_kernel<<<1, 256, 0, stream>>>(stats, 256);
      bn_reduce_kernel<<<512, 128, 0, stream>>>(bufA, stats, N);
      // normalize h in place, dual-write into agg for the next layer
      bn_apply_kernel<<<applyBlocks, 256, 0, stream>>>(
          bufA, bufB, stats, bn_g + l * H, bn_b + l * H, N);
    }
  }

  // post_mp
  gemm_wmma_kernel<128, true, false>
      <<<mtiles, 256, 0, stream>>>(bufA, post_W1, post_b1, bufC, nullptr, N);
  post2_kernel<<<(N * 8 + 255) / 256, 256, 0, stream>>>(bufC, post_W2, post_b2,
                                                        out, N);
}